// test_model_5334349381894
// MI455X (gfx1250) — compile-verified
//
#include <hip/hip_runtime.h>
#include <stdint.h>

// FPS: B=8 batches, N=131072 points (xyz f32), NPOINT=1024 samples.
// 8 blocks per batch; each block stages its 16384-point slab (192KB) into LDS
// once via the Tensor Data Mover, then runs 1024 sequential FPS iterations with
// register-resident mindist and an L2-atomic cross-block argmax exchange.

#define BATCHES 8
#define NPTS    131072
#define NPOINT  1024
#define KBLK    8                    // blocks per batch
#define WGS     1024                 // threads per block (32 waves, wave32)
#define SLAB    (NPTS / KBLK)        // 16384 points per block
#define PPT     (SLAB / WGS)         // 16 points per thread

typedef unsigned int       u32;
typedef unsigned long long u64;

typedef unsigned int v4u __attribute__((ext_vector_type(4)));
typedef int          v8i __attribute__((ext_vector_type(8)));
typedef int          v4i __attribute__((ext_vector_type(4)));

__global__ void fps_init_ws(u32* ws) {
    int t = threadIdx.x;
    if (t < 512) ws[t] = 0u;   // 256 dwords counters (128B stride/batch) + 256 dwords key slots
}

__launch_bounds__(WGS, 1)
__global__ void fps_kernel(const float* __restrict__ in1,
                           float* __restrict__ out,
                           u32* __restrict__ ws) {
    extern __shared__ float smem[];
    float* sxyz = smem;                         // SLAB*3 floats = 192KB
    u64*   swk  = (u64*)(smem + SLAB * 3);      // 32 per-wave keys
    float* bc   = (float*)(swk + 32);           // centroid broadcast (3 floats)

    const int tid   = threadIdx.x;
    const int b     = blockIdx.x >> 3;
    const int kb    = blockIdx.x & 7;
    const int gbase = kb * SLAB;

    const float* batch = in1 + (size_t)b * NPTS * 3;
    u32* counter = ws + b * 32;                       // 128B apart per batch
    u64* keysAll = (u64*)((char*)ws + 1024);          // [parity][batch][kblk]

    // ---- Stage this block's coordinate slab into LDS via the TDM (wave 0 issues) ----
    if (tid < 32) {
        uint64_t ga = (uint64_t)(uintptr_t)(batch + (size_t)gbase * 3);
        u32 lds_off = (u32)(uintptr_t)sxyz;  // low 32 bits of flat shared ptr = LDS offset
        v4u g0;
        v8i g1;
        v4i g2 = {0, 0, 0, 0};
        v4i g3 = {0, 0, 0, 0};
        v8i g4 = {0, 0, 0, 0, 0, 0, 0, 0};
        g0.x = 1u;                                             // count=1, user descriptor
        g0.y = lds_off;                                        // lds_addr
        g0.z = (u32)(ga & 0xFFFFFFFFu);                        // global_addr[31:0]
        g0.w = ((u32)(ga >> 32) & 0x01FFFFFFu) | 0x80000000u;  // global_addr[56:32] | type=2
        // 2D tile: 4096-dword rows x 12 rows = 49152 dwords = 192KB, contiguous
        g1[0] = (int)(2u << 16);                 // data_size = 4 bytes
        g1[1] = (int)(4096u << 16);              // tensor_dim0 = 4096
        g1[2] = (int)(12u << 16);                // tensor_dim1 = 12
        g1[3] = (int)(4096u << 16);              // tile_dim0 = 4096
        g1[4] = (int)12u;                        // tile_dim1 = 12, tile_dim2 = 0
        g1[5] = (int)4096u;                      // tensor_dim0_stride = 4096
        g1[6] = (int)((49152u & 0xFFFFu) << 16); // tensor_dim1_stride lo16 (unused, 2D)
        g1[7] = 0;
        __builtin_amdgcn_tensor_load_to_lds(g0, g1, g2, g3, g4, 0);
        __builtin_amdgcn_s_wait_tensorcnt(0);
    }
    if (tid == 0) {
        __builtin_prefetch(batch, 0, 2);          // global_prefetch_b8
        bc[0] = batch[0];                         // first centroid = point 0
        bc[1] = batch[1];
        bc[2] = batch[2];
    }
    __syncthreads();

    float md[PPT];
#pragma unroll
    for (int j = 0; j < PPT; ++j) md[j] = __builtin_huge_valf();

    for (int it = 0; it < NPOINT; ++it) {
        float cx = bc[0], cy = bc[1], cz = bc[2];

        // out[b, it, :] = centroid used this iteration (idx[0] == 0)
        if (kb == 0 && tid == 0) {
            float* o = out + ((size_t)b * NPOINT + it) * 3;
            o[0] = cx; o[1] = cy; o[2] = cz;
        }

        u64 best = 0;
#pragma unroll
        for (int j = 0; j < PPT; ++j) {
            int lp = tid + j * WGS;
            float x = sxyz[lp * 3 + 0];
            float y = sxyz[lp * 3 + 1];
            float z = sxyz[lp * 3 + 2];
            float dx = x - cx, dy = y - cy, dz = z - cz;
            float d = __builtin_fmaf(dz, dz, __builtin_fmaf(dy, dy, dx * dx));
            float m = fminf(md[j], d);
            md[j] = m;
            // dist >= 0 so float bits are order-monotonic; ~idx gives lowest-index tie-break
            u64 key = ((u64)__float_as_uint(m) << 32) | (u32)(~(u32)(gbase + lp));
            best = key > best ? key : best;
        }

        // wave32 reduction
#pragma unroll
        for (int off = 16; off > 0; off >>= 1) {
            u64 o = __shfl_xor(best, off, 32);
            best = o > best ? o : best;
        }
        if ((tid & 31) == 0) swk[tid >> 5] = best;
        __syncthreads();

        if (tid < 32) {
            best = swk[tid];
#pragma unroll
            for (int off = 16; off > 0; off >>= 1) {
                u64 o = __shfl_xor(best, off, 32);
                best = o > best ? o : best;
            }
            if (tid == 0) {
                // cross-block exchange for this batch (parity double-buffered slots)
                u64* keys = keysAll + (size_t)(it & 1) * (BATCHES * KBLK) + (size_t)b * KBLK;
                __hip_atomic_store(&keys[kb], best, __ATOMIC_RELAXED, __HIP_MEMORY_SCOPE_AGENT);
                __hip_atomic_fetch_add(counter, 1u, __ATOMIC_RELEASE, __HIP_MEMORY_SCOPE_AGENT);
                u32 target = (u32)(KBLK * (it + 1));   // monotonic phase counter
                while (__hip_atomic_load(counter, __ATOMIC_ACQUIRE, __HIP_MEMORY_SCOPE_AGENT) < target)
                    __builtin_amdgcn_s_sleep(1);
                u64 w = 0;
#pragma unroll
                for (int q = 0; q < KBLK; ++q) {
                    u64 v = __hip_atomic_load(&keys[q], __ATOMIC_RELAXED, __HIP_MEMORY_SCOPE_AGENT);
                    w = v > w ? v : w;
                }
                u32 widx = ~(u32)w;                    // winning global point index
                const float* cp = batch + (size_t)widx * 3;
                bc[0] = cp[0]; bc[1] = cp[1]; bc[2] = cp[2];
            }
        }
        __syncthreads();
    }
}

extern "C" void kernel_launch(void* const* d_in, const int* in_sizes, int n_in,
                              void* d_out, int out_size, void* d_ws, size_t ws_size,
                              hipStream_t stream) {
    (void)in_sizes; (void)n_in; (void)out_size; (void)ws_size;
    const float* in1 = (const float*)d_in[0];   // [8, 131072, 3] f32
    float* out = (float*)d_out;                 // [8, 1024, 3] f32
    u32* ws = (u32*)d_ws;

    size_t smem = (size_t)SLAB * 3 * sizeof(float) + 32 * sizeof(u64) + 4 * sizeof(float);
    (void)hipFuncSetAttribute((const void*)fps_kernel,
                              hipFuncAttributeMaxDynamicSharedMemorySize, (int)smem);

    fps_init_ws<<<1, 512, 0, stream>>>(ws);
    fps_kernel<<<BATCHES * KBLK, WGS, smem, stream>>>(in1, out, ws);
}